// CriticSwapGNN_79860621902162
// MI455X (gfx1250) — compile-verified
//
#include <hip/hip_runtime.h>
#include <math.h>

typedef __attribute__((ext_vector_type(2))) float v2f;
typedef __attribute__((ext_vector_type(8))) float v8f;

#define NNODES 100000
#define NEDGES 1600000
#define NB     8
#define HID    64
#define NHEAD  4
#define CH     16
#define NUM_LOC 15
#define K1PAD  68   // layer-1 input features padded 66 -> 68 (multiple of 4)

// ---------------------------------------------------------------------------
// small helpers
// ---------------------------------------------------------------------------
__device__ __forceinline__ float lrelu02(float v) { return v > 0.f ? v : 0.2f * v; }

__device__ __forceinline__ void atomicMaxF(float* addr, float v) {
  // monotonic bit trick: int-max for non-negative, uint-min for negative
  if (v >= 0.f) atomicMax((int*)addr, __float_as_int(v));
  else          atomicMin((unsigned int*)addr, __float_as_uint(v));
}

// ---------------------------------------------------------------------------
// 0) zero stats / pool accumulators
// ---------------------------------------------------------------------------
__global__ void k_init_stats(float* stats) {
  int t = threadIdx.x;
  if (t < 64) stats[t] = 0.f;
}

// ---------------------------------------------------------------------------
// 1) sum / sumsq of requests[NUM_LOC:]
// ---------------------------------------------------------------------------
__global__ void k_reduce_tail(const float* __restrict__ req, int n, float* stats) {
  __shared__ float ssum[256], ssq[256];
  int tid = threadIdx.x;
  float s = 0.f, q = 0.f;
  for (int i = blockIdx.x * blockDim.x + tid + NUM_LOC; i < n;
       i += gridDim.x * blockDim.x) {
    float v = req[i];
    s += v; q += v * v;
  }
  ssum[tid] = s; ssq[tid] = q;
  __syncthreads();
  for (int off = 128; off > 0; off >>= 1) {
    if (tid < off) { ssum[tid] += ssum[tid + off]; ssq[tid] += ssq[tid + off]; }
    __syncthreads();
  }
  if (tid == 0) {
    atomicAdd(&stats[0], ssum[0]);
    atomicAdd(&stats[1], ssq[0]);
  }
}

// ---------------------------------------------------------------------------
// 2) build padded input features: xpad[N][68]
// ---------------------------------------------------------------------------
__global__ void k_build_x_main(const int* __restrict__ type_ids,
                               const float* __restrict__ emb,
                               float* __restrict__ xpad, int n) {
  long long tid = (long long)blockIdx.x * blockDim.x + threadIdx.x;
  if (tid >= (long long)n * 64) return;
  int i = (int)(tid >> 6), j = (int)(tid & 63);
  xpad[(size_t)i * K1PAD + j] = emb[type_ids[i] * 64 + j];
}

__global__ void k_build_x_tail(const float* __restrict__ req,
                               const float* __restrict__ upd,
                               const float* __restrict__ stats,
                               float* __restrict__ xpad, int n) {
  int i = blockIdx.x * blockDim.x + threadIdx.x;
  if (i >= n) return;
  float m = (float)(n - NUM_LOC);
  float mean = stats[0] / m;
  float var  = (stats[1] - stats[0] * stats[0] / m) / (m - 1.f);
  float inv  = 1.f / (sqrtf(var) + 1e-6f);
  float r = req[i];
  if (i >= NUM_LOC) r = (r - mean) * inv;
  float* row = xpad + (size_t)i * K1PAD;
  row[64] = r;
  row[65] = upd[i];
  row[66] = 0.f;
  row[67] = 0.f;
}

// ---------------------------------------------------------------------------
// 3) WMMA GEMM: Y[n][64] = act( X[n][kpad(=ldx)] @ W[kreal][64] + bias )
//    block = 256 threads = 8 waves; wave owns 16 rows x 64 cols.
//    f32 WMMA 16x16x4, W staged in LDS (zero padded to kpad rows).
// ---------------------------------------------------------------------------
__global__ __launch_bounds__(256) void k_gemm_n64(
    const float* __restrict__ X, int ldx, int kreal,
    const float* __restrict__ W, const float* __restrict__ bias, int relu,
    float* __restrict__ Y, int nrows) {
  __shared__ float Wlds[K1PAD * 64];
  const int tid = threadIdx.x;
  const int kpad = ldx;
  for (int idx = tid; idx < kpad * 64; idx += 256) {
    int k = idx >> 6, nn = idx & 63;
    Wlds[idx] = (k < kreal) ? W[k * 64 + nn] : 0.f;
  }
  __syncthreads();

  const int wid = tid >> 5, lane = tid & 31;
  const int m0 = blockIdx.x * 128 + wid * 16;
  if (m0 >= nrows) return;  // wave-uniform: EXEC stays all-ones for WMMA

  const int mrow = lane & 15;            // A-matrix M index
  const int koff = (lane >> 4) << 1;     // lanes 16-31 hold K+2,K+3
  const int ncol = lane & 15;            // B/C/D N index within tile

  v8f acc[4] = {{0}, {0}, {0}, {0}};
  const float* xbase = X + (size_t)(m0 + mrow) * ldx + koff;

  for (int k0 = 0; k0 < kpad; k0 += 4) {
    v2f a = *(const v2f*)(xbase + k0);
    const float* w0 = &Wlds[(k0 + koff) * 64 + ncol];
#pragma unroll
    for (int t = 0; t < 4; ++t) {
      v2f b;
      b.x = w0[t * 16];
      b.y = w0[64 + t * 16];
      acc[t] = __builtin_amdgcn_wmma_f32_16x16x4_f32(
          false, a, false, b, (short)0, acc[t], false, false);
    }
  }

  const int rowoff = (lane >> 4) << 3;  // lanes 16-31 -> rows +8
#pragma unroll
  for (int t = 0; t < 4; ++t) {
    int col = t * 16 + ncol;
    float bv = bias ? bias[col] : 0.f;
#pragma unroll
    for (int v = 0; v < 8; ++v) {
      int row = m0 + v + rowoff;
      float val = acc[t][v] + bv;
      if (relu) val = fmaxf(val, 0.f);
      Y[(size_t)row * 64 + col] = val;
    }
  }
}

// ---------------------------------------------------------------------------
// 4) per-(node,head) attention logits s_src / s_dst
// ---------------------------------------------------------------------------
__global__ void k_scores(const float* __restrict__ H,
                         const float* __restrict__ a_src,
                         const float* __restrict__ a_dst,
                         float* __restrict__ s_src, float* __restrict__ s_dst,
                         int n) {
  int tid = blockIdx.x * blockDim.x + threadIdx.x;
  if (tid >= n * NHEAD) return;
  int i = tid >> 2, h = tid & 3;
  const float* hp = H + (size_t)i * 64 + h * CH;
  const float* as = a_src + h * CH;
  const float* ad = a_dst + h * CH;
  float ss = 0.f, sd = 0.f;
#pragma unroll
  for (int c = 0; c < CH; ++c) { ss += hp[c] * as[c]; sd += hp[c] * ad[c]; }
  s_src[tid] = ss;
  s_dst[tid] = sd;
}

// ---------------------------------------------------------------------------
// 5) reset per-layer accumulators: out=0, mmax=-inf, zsum=0
// ---------------------------------------------------------------------------
__global__ void k_layer_init(float* __restrict__ out, float* __restrict__ mmax,
                             float* __restrict__ zsum, int n) {
  long long tid = (long long)blockIdx.x * blockDim.x + threadIdx.x;
  if (tid < (long long)n * 64) out[tid] = 0.f;
  if (tid < (long long)n * NHEAD) {
    mmax[tid] = -__builtin_huge_valf();
    zsum[tid] = 0.f;
  }
}

// ---------------------------------------------------------------------------
// 6) edge pass 1: segment max of leaky-relu logits (scores recomputed)
// ---------------------------------------------------------------------------
__global__ void k_edge_max(const int* __restrict__ src, const int* __restrict__ dst,
                           const float* __restrict__ s_src,
                           const float* __restrict__ s_dst,
                           float* __restrict__ mmax, int nedge, int n) {
  int e = blockIdx.x * blockDim.x + threadIdx.x;
  if (e >= nedge + n) return;
  int s, d;
  if (e < nedge) { s = src[e]; d = dst[e]; } else { s = d = e - nedge; }
#pragma unroll
  for (int h = 0; h < NHEAD; ++h) {
    float v = lrelu02(s_src[s * NHEAD + h] + s_dst[d * NHEAD + h]);
    atomicMaxF(&mmax[d * NHEAD + h], v);
  }
}

// ---------------------------------------------------------------------------
// 7) edge pass 2: segment sum of exp(e - max)
// ---------------------------------------------------------------------------
__global__ void k_edge_sum(const int* __restrict__ src, const int* __restrict__ dst,
                           const float* __restrict__ s_src,
                           const float* __restrict__ s_dst,
                           const float* __restrict__ mmax,
                           float* __restrict__ zsum, int nedge, int n) {
  int e = blockIdx.x * blockDim.x + threadIdx.x;
  if (e >= nedge + n) return;
  int s, d;
  if (e < nedge) { s = src[e]; d = dst[e]; } else { s = d = e - nedge; }
#pragma unroll
  for (int h = 0; h < NHEAD; ++h) {
    float v = lrelu02(s_src[s * NHEAD + h] + s_dst[d * NHEAD + h]);
    atomicAdd(&zsum[d * NHEAD + h], __expf(v - mmax[d * NHEAD + h]));
  }
}

// ---------------------------------------------------------------------------
// 8) edge pass 3: one wave32 per edge; lane L covers channels L and L+32.
//    coalesced 128B gathers of h[src] and 128B atomic bursts into out[dst].
// ---------------------------------------------------------------------------
__global__ __launch_bounds__(256) void k_edge_agg(
    const int* __restrict__ src, const int* __restrict__ dst,
    const float* __restrict__ H, const float* __restrict__ s_src,
    const float* __restrict__ s_dst, const float* __restrict__ mmax,
    const float* __restrict__ zsum, float* __restrict__ out,
    int nedge, int n) {
  int wid = threadIdx.x >> 5, lane = threadIdx.x & 31;
  int e = blockIdx.x * 8 + wid;
  if (e >= nedge + n) return;
  int s, d;
  if (e < nedge) { s = src[e]; d = dst[e]; } else { s = d = e - nedge; }

  int h1 = lane >> 4;        // head for channel = lane
  int h2 = 2 + (lane >> 4);  // head for channel = lane + 32
  float v1 = lrelu02(s_src[s * NHEAD + h1] + s_dst[d * NHEAD + h1]);
  float v2 = lrelu02(s_src[s * NHEAD + h2] + s_dst[d * NHEAD + h2]);
  float a1 = __expf(v1 - mmax[d * NHEAD + h1]) / (zsum[d * NHEAD + h1] + 1e-16f);
  float a2 = __expf(v2 - mmax[d * NHEAD + h2]) / (zsum[d * NHEAD + h2] + 1e-16f);

  const float* hs = H + (size_t)s * 64;
  float* od = out + (size_t)d * 64;
  atomicAdd(&od[lane],      hs[lane]      * a1);
  atomicAdd(&od[lane + 32], hs[lane + 32] * a2);
}

// ---------------------------------------------------------------------------
// 9) bias + optional relu: X = act(out + b)
// ---------------------------------------------------------------------------
__global__ void k_bias_act(const float* __restrict__ in,
                           const float* __restrict__ bias, int relu,
                           float* __restrict__ outx, int n) {
  long long tid = (long long)blockIdx.x * blockDim.x + threadIdx.x;
  if (tid >= (long long)n * 64) return;
  float v = in[tid] + bias[tid & 63];
  if (relu) v = fmaxf(v, 0.f);
  outx[tid] = v;
}

// ---------------------------------------------------------------------------
// 10) node_value = x . out_W + out_b  ; segment-sum into batch pools
// ---------------------------------------------------------------------------
__global__ void k_out_pool(const float* __restrict__ X,
                           const float* __restrict__ outW, const float* __restrict__ outB,
                           const int* __restrict__ batch,
                           float* __restrict__ stats, int n) {
  int i = blockIdx.x * blockDim.x + threadIdx.x;
  if (i >= n) return;
  const float* xr = X + (size_t)i * 64;
  float v = outB[0];
#pragma unroll
  for (int j = 0; j < 64; ++j) v += xr[j] * outW[j];
  int b = batch[i];
  atomicAdd(&stats[16 + b], v);   // numerator
  atomicAdd(&stats[24 + b], 1.f); // count
}

__global__ void k_finalize(const float* __restrict__ stats, float* __restrict__ out) {
  int b = threadIdx.x;
  if (b < NB) out[b] = stats[16 + b] / stats[24 + b];
}

// ---------------------------------------------------------------------------
// launcher
// ---------------------------------------------------------------------------
extern "C" void kernel_launch(void* const* d_in, const int* in_sizes, int n_in,
                              void* d_out, int out_size, void* d_ws, size_t ws_size,
                              hipStream_t stream) {
  const int*   type_ids = (const int*)  d_in[0];
  const float* upd      = (const float*)d_in[1];
  const float* req      = (const float*)d_in[2];
  /* latency d_in[3] unused by reference */
  const int*   eidx     = (const int*)  d_in[4];
  const int*   batch    = (const int*)  d_in[5];
  const float* emb      = (const float*)d_in[6];
  const float* W1       = (const float*)d_in[7];
  const float* as1      = (const float*)d_in[8];
  const float* ad1      = (const float*)d_in[9];
  const float* b1       = (const float*)d_in[10];
  const float* Wh       = (const float*)d_in[11];
  const float* ash      = (const float*)d_in[12];
  const float* adh      = (const float*)d_in[13];
  const float* bh       = (const float*)d_in[14];
  const float* mlpW     = (const float*)d_in[15];
  const float* mlpB     = (const float*)d_in[16];
  const float* outW     = (const float*)d_in[17];
  const float* outB     = (const float*)d_in[18];

  const int N = in_sizes[0];
  const int E = in_sizes[4] / 2;
  const int EN = E + N;
  const int* src = eidx;
  const int* dst = eidx + E;

  // workspace carve-up (floats)
  float* ws    = (float*)d_ws;
  float* stats = ws;                                // 64 floats
  float* xpad  = stats + 64;                        // N*68
  float* bufX  = xpad + (size_t)N * K1PAD;          // N*64
  float* bufH  = bufX + (size_t)N * 64;             // N*64
  float* bufO  = bufH + (size_t)N * 64;             // N*64
  float* s_src = bufO + (size_t)N * 64;             // N*4
  float* s_dst = s_src + (size_t)N * NHEAD;         // N*4
  float* mmax  = s_dst + (size_t)N * NHEAD;         // N*4
  float* zsum  = mmax  + (size_t)N * NHEAD;         // N*4

  const int TB = 256;
  auto cdiv = [](long long a, long long b) { return (int)((a + b - 1) / b); };

  // stage 0: stats, input features
  k_init_stats<<<1, 64, 0, stream>>>(stats);
  k_reduce_tail<<<512, TB, 0, stream>>>(req, N, stats);
  k_build_x_main<<<cdiv((long long)N * 64, TB), TB, 0, stream>>>(type_ids, emb, xpad, N);
  k_build_x_tail<<<cdiv(N, TB), TB, 0, stream>>>(req, upd, stats, xpad, N);

  const int gemm_blocks = cdiv(N, 128);
  const int nh_blocks   = cdiv((long long)N * 64, TB);
  const int en_blocks   = cdiv(EN, TB);
  const int agg_blocks  = cdiv(EN, 8);

  // 4 GAT layers (layer 0 reads xpad/K=66, layers 1..3 read bufX/K=64)
  for (int layer = 0; layer < 4; ++layer) {
    const float* gin  = (layer == 0) ? xpad : bufX;
    int ldx   = (layer == 0) ? K1PAD : 64;
    int kreal = (layer == 0) ? 66 : 64;
    const float* Wl = (layer == 0) ? W1 : Wh  + (size_t)(layer - 1) * 64 * 64;
    const float* al = (layer == 0) ? as1 : ash + (size_t)(layer - 1) * NHEAD * CH;
    const float* dl = (layer == 0) ? ad1 : adh + (size_t)(layer - 1) * NHEAD * CH;
    const float* bl = (layer == 0) ? b1  : bh  + (size_t)(layer - 1) * 64;
    int relu = (layer < 3) ? 1 : 0;

    k_gemm_n64<<<gemm_blocks, TB, 0, stream>>>(gin, ldx, kreal, Wl, nullptr, 0, bufH, N);
    k_scores<<<cdiv((long long)N * NHEAD, TB), TB, 0, stream>>>(bufH, al, dl, s_src, s_dst, N);
    k_layer_init<<<nh_blocks, TB, 0, stream>>>(bufO, mmax, zsum, N);
    k_edge_max<<<en_blocks, TB, 0, stream>>>(src, dst, s_src, s_dst, mmax, E, N);
    k_edge_sum<<<en_blocks, TB, 0, stream>>>(src, dst, s_src, s_dst, mmax, zsum, E, N);
    k_edge_agg<<<agg_blocks, TB, 0, stream>>>(src, dst, bufH, s_src, s_dst, mmax, zsum, bufO, E, N);
    k_bias_act<<<nh_blocks, TB, 0, stream>>>(bufO, bl, relu, bufX, N);
  }

  // 3 MLP layers, bias+relu fused into the WMMA GEMM epilogue
  k_gemm_n64<<<gemm_blocks, TB, 0, stream>>>(bufX, 64, 64, mlpW,             mlpB,      1, bufH, N);
  k_gemm_n64<<<gemm_blocks, TB, 0, stream>>>(bufH, 64, 64, mlpW + 64 * 64,   mlpB + 64, 1, bufO, N);
  k_gemm_n64<<<gemm_blocks, TB, 0, stream>>>(bufO, 64, 64, mlpW + 2 * 64 * 64, mlpB + 128, 1, bufX, N);

  // output projection + batch mean pooling
  k_out_pool<<<cdiv(N, TB), TB, 0, stream>>>(bufX, outW, outB, batch, stats, N);
  k_finalize<<<1, 64, 0, stream>>>(stats, (float*)d_out);
}